// Loss_39341900431615
// MI455X (gfx1250) — compile-verified
//
#include <hip/hip_runtime.h>

// ---------------------------------------------------------------------------
// out[w] = sum_{j,k} ((x0-j)^2 + (y0-k)^2) * T[j,k,w],  T = tensor[0] (f32,
// [128,128,128], w contiguous), (x0,y0) = argmax position of tensor[0,0,:,:].
// Bandwidth-bound GEMV (8.4 MB @ 23.3 TB/s ~ 0.36us); executed with
// V_WMMA_F32_16X16X4_F32 (broadcast weight row in A, streaming B tiles).
// ---------------------------------------------------------------------------

typedef __attribute__((ext_vector_type(2))) float v2f;
typedef __attribute__((ext_vector_type(8))) float v8f;

#define HC 128
#define WC 128
#define JK_TOTAL (HC * WC)   // 16384

// Kernel 1: argmax of tensor[0,0,:,:] (first occurrence, row-major),
// write x0,y0 (as floats) into ws[0..1], and zero d_out[0..127].
__global__ void loss_argmax_init(const float* __restrict__ t,
                                 float* __restrict__ ws,
                                 float* __restrict__ out) {
  __shared__ float sval[256];
  __shared__ int   sidx[256];
  const int tid = threadIdx.x;

  float best = -__builtin_inff();
  int   bidx = 0;
  for (int i = tid; i < JK_TOTAL; i += 256) {
    float v = t[i];
    if (v > best) { best = v; bidx = i; }   // strict >: keeps lowest i per thread
  }
  sval[tid] = best;
  sidx[tid] = bidx;
  __syncthreads();

  for (int s = 128; s > 0; s >>= 1) {
    if (tid < s) {
      float v2 = sval[tid + s];
      int   i2 = sidx[tid + s];
      if (v2 > sval[tid] || (v2 == sval[tid] && i2 < sidx[tid])) {
        sval[tid] = v2;
        sidx[tid] = i2;
      }
    }
    __syncthreads();
  }

  if (tid == 0) {
    int idx = sidx[0];
    ws[0] = (float)(idx / WC);   // x0
    ws[1] = (float)(idx % WC);   // y0
  }
  if (tid < WC) out[tid] = 0.0f;   // zero accumulator (harness poisons d_out)
}

// Kernel 2: weighted reduction via f32 WMMA.
// grid = (8 w-tiles, 8 slice-groups), block = 256 (8 waves).
// Each wave: one 16-wide w tile x one 256-row jk slice -> 64 WMMAs.
__global__ void loss_wmma_reduce(const float* __restrict__ t,
                                 const float* __restrict__ ws,
                                 float* __restrict__ out) {
  const int lane   = threadIdx.x & 31;
  const int wave   = threadIdx.x >> 5;
  const int n      = lane & 15;           // N column within tile
  const int kkbase = (lane >> 4) << 1;    // 0 for lanes 0-15, 2 for lanes 16-31

  const int w0       = blockIdx.x * 16;          // 8 tiles cover W=128
  const int slice    = blockIdx.y * 8 + wave;    // 0..63
  const int jk_begin = slice * 256;              // 256 jk rows per wave

  const float x0 = ws[0];
  const float y0 = ws[1];

  // B element pointer: T[(jk_begin + kkbase)][w0 + n]; +WC gives next K row.
  const float* p = t + (size_t)(jk_begin + kkbase) * WC + w0 + n;

  v8f c = {};
#pragma unroll 4
  for (int it = 0; it < 64; ++it) {
    const int jk = jk_begin + it * 4 + kkbase;

    // A = weight row broadcast over all 16 M rows (depends only on K).
    const int j0 = jk >> 7,        k0 = jk & 127;
    const int j1 = (jk + 1) >> 7,  k1 = (jk + 1) & 127;
    const float dx0 = x0 - (float)j0, dy0 = y0 - (float)k0;
    const float dx1 = x0 - (float)j1, dy1 = y0 - (float)k1;
    v2f a;
    a.x = dx0 * dx0 + dy0 * dy0;
    a.y = dx1 * dx1 + dy1 * dy1;

    // B tile: two coalesced dword loads per lane (rows 512 B apart).
    v2f b;
    b.x = p[0];
    b.y = p[WC];
    p += 4 * WC;

    // D = A x B + C   (accumulate in registers)
    c = __builtin_amdgcn_wmma_f32_16x16x4_f32(
        /*neg_a=*/false, a, /*neg_b=*/false, b,
        /*c_mod=*/(short)0, c, /*reuse_a=*/false, /*reuse_b=*/false);
  }

  // All D rows identical; VGPR0 on lanes 0-15 is D[0][n] = partial[w0+n].
  if (lane < 16) {
    atomicAdd(&out[w0 + lane], c[0]);
  }
}

extern "C" void kernel_launch(void* const* d_in, const int* in_sizes, int n_in,
                              void* d_out, int out_size, void* d_ws, size_t ws_size,
                              hipStream_t stream) {
  (void)in_sizes; (void)n_in; (void)out_size; (void)ws_size;
  const float* t   = (const float*)d_in[0];
  float*       out = (float*)d_out;
  float*       ws  = (float*)d_ws;   // ws[0]=x0, ws[1]=y0

  loss_argmax_init<<<1, 256, 0, stream>>>(t, ws, out);

  dim3 grid(8, 8);   // 8 w-tiles x 8 slice groups (8 waves each) = 512 waves
  loss_wmma_reduce<<<grid, 256, 0, stream>>>(t, ws, out);
}